// IFNet_46385646797617
// MI455X (gfx1250) — compile-verified
//
#include <hip/hip_runtime.h>
#include <math.h>

// Problem constants (B=2, C=3, H=W=384, 18 unfolded channels x 9 taps = 162)
#define HH   384
#define WW   384
#define HWP  (HH * WW)
#define KD   162
#define TX   64
#define TY   4
#define LW   (TX + 2)      // 66 tile width with halo
#define LHH  (TY + 2)      // 6  tile height with halo
#define LCH  (LW * LHH)    // 396 elements per channel tile
#define FUSE_ELEMS (2 * 3 * HWP)

// ---- CDNA5 async global->LDS path ------------------------------------------
#if defined(__HIP_DEVICE_COMPILE__) && defined(__has_builtin)
#if __has_builtin(__builtin_amdgcn_global_load_async_to_lds_b32)
#define ASYNC_BUILTIN 1
#endif
#if __has_builtin(__builtin_amdgcn_s_wait_asynccnt)
#define WAITASYNC_BUILTIN 1
#endif
#endif

typedef __attribute__((address_space(1))) int gas_int;
typedef __attribute__((address_space(3))) int lds_int;
typedef __attribute__((address_space(3))) char lds_char;

__device__ __forceinline__ void async_copy_b32(const float* g, float* l) {
#if defined(ASYNC_BUILTIN)
    // signature: (addrspace(1) int* src, addrspace(3) int* dst, imm offset, imm cpol)
    __builtin_amdgcn_global_load_async_to_lds_b32(
        (gas_int*)g, (lds_int*)l, 0, 0);
#else
    unsigned int ldsoff = (unsigned int)(unsigned long long)(lds_char*)l;
    asm volatile("global_load_async_to_lds_b32 %0, %1, off"
                 :: "v"(ldsoff), "v"(g) : "memory");
#endif
}

__device__ __forceinline__ void wait_async0() {
#if defined(WAITASYNC_BUILTIN)
    __builtin_amdgcn_s_wait_asynccnt(0);
#else
    asm volatile("s_wait_asynccnt 0x0" ::: "memory");
#endif
}
// ----------------------------------------------------------------------------

__global__ __launch_bounds__(TX * TY) void fuse_softmax_kernel(
    const float* __restrict__ img,     // [2,3,384,384] deshadow_img
    const float* __restrict__ pp,      // [2,6] pred_param
    const float* __restrict__ kern,    // [2,486,384,384]
    float* __restrict__ out)           // fuse_img (884736) ++ dark_img (884736)
{
    __shared__ float s_img[3 * LCH];   // raw image tile + halo, 4.75 KB

    const int lx  = threadIdx.x;
    const int ly  = threadIdx.y;
    const int tid = ly * TX + lx;
    const int b   = blockIdx.z;
    const int x0  = blockIdx.x * TX - 1;   // halo origin
    const int y0  = blockIdx.y * TY - 1;

    const float* imgB = img + (size_t)b * 3 * HWP;

    // Stage 66x6x3 raw-image tile into LDS via async global->LDS copies.
    // OOB halo entries get raw=-1 so d01 = 0.5*x+0.5 == 0 (zero padding).
    for (int idx = tid; idx < 3 * LCH; idx += TX * TY) {
        int ch = idx / LCH;
        int r  = idx - ch * LCH;
        int yy = r / LW;
        int xx = r - yy * LW;
        int gx = x0 + xx;
        int gy = y0 + yy;
        if (gx >= 0 && gx < WW && gy >= 0 && gy < HH) {
            async_copy_b32(imgB + (size_t)ch * HWP + gy * WW + gx, &s_img[idx]);
        } else {
            s_img[idx] = -1.0f;
        }
    }
    wait_async0();
    __syncthreads();

    const int gx = x0 + 1 + lx;            // in-range: 384 % 64 == 0
    const int gy = y0 + 1 + ly;            // in-range: 384 % 4  == 0
    const int p  = gy * WW + gx;

    // 27 neighbor d01 values in registers (conflict-free LDS reads).
    float nbr[3][9];
#pragma unroll
    for (int cc = 0; cc < 3; ++cc)
#pragma unroll
        for (int i = 0; i < 3; ++i)
#pragma unroll
            for (int j = 0; j < 3; ++j)
                nbr[cc][i * 3 + j] =
                    fmaf(0.5f, s_img[cc * LCH + (ly + i) * LW + (lx + j)], 0.5f);

    // Per-batch scale factors for the 18 concat channels.
    float mean[3], minv[3];
#pragma unroll
    for (int cc = 0; cc < 3; ++cc) {
        mean[cc] = pp[b * 6 + 2 * cc];     // pred_param[:,0,2,4]
        minv[cc] = pp[b * 6 + 2 * cc + 1]; // pred_param[:,1,3,5]
    }
    float fac[18];
#pragma unroll
    for (int cc = 0; cc < 3; ++cc) {
        const float b2 = (1.0f - mean[cc]) * (1.0f / 3.0f);
        fac[cc]              = 1.0f;                        // d01 itself
        fac[3 + 0 * 3 + cc]  = minv[cc];                    // scale 0
        fac[3 + 1 * 3 + cc]  = 0.5f * (minv[cc] + mean[cc]);// scale 1
        fac[3 + 2 * 3 + cc]  = mean[cc];                    // scale 2
        fac[3 + 3 * 3 + cc]  = mean[cc] + b2;               // scale 3
        fac[3 + 4 * 3 + cc]  = mean[cc] + 2.0f * b2;        // scale 4
    }

    // Fused single-pass softmax-dot over 162 taps per output channel.
    const float* kb = kern + (size_t)b * 3 * KD * HWP + p;
#pragma unroll 1
    for (int c = 0; c < 3; ++c) {
        const float* pk = kb + (size_t)c * KD * HWP;
        float m = -__builtin_inff();
        float s = 0.0f;
        float d = 0.0f;
#pragma unroll
        for (int cb = 0; cb < 6; ++cb) {        // 6 chunks x 27 taps
            float w[27];
#pragma unroll
            for (int j = 0; j < 27; ++j)        // NT: 573MB stream, no reuse
                w[j] = __builtin_nontemporal_load(pk + (size_t)(cb * 27 + j) * HWP);
            float cm = w[0];
#pragma unroll
            for (int j = 1; j < 27; ++j) cm = fmaxf(cm, w[j]);
            const float nm = fmaxf(m, cm);
            const float rescale = __expf(m - nm);   // 0 on first chunk
            s *= rescale;
            d *= rescale;
#pragma unroll
            for (int j = 0; j < 27; ++j) {
                const int ch  = 3 * cb + j / 9;
                const int tap = j - (j / 9) * 9;
                const int cc  = (ch < 3) ? ch : (ch - 3) % 3;
                const float e = __expf(w[j] - nm);
                s += e;
                d = fmaf(e, fac[ch] * nbr[cc][tap], d);
            }
            m = nm;
        }
        const float fuse = d / s;
        out[(size_t)(b * 3 + c) * HWP + p] = fmaf(2.0f, fuse, -1.0f);
    }

    // dark_img = d01*mean*2 - 1 = (raw+1)*mean - 1
#pragma unroll
    for (int cc = 0; cc < 3; ++cc) {
        const float raw = s_img[cc * LCH + (ly + 1) * LW + (lx + 1)];
        out[FUSE_ELEMS + (size_t)(b * 3 + cc) * HWP + p] =
            fmaf(raw + 1.0f, mean[cc], -1.0f);
    }
}

extern "C" void kernel_launch(void* const* d_in, const int* in_sizes, int n_in,
                              void* d_out, int out_size, void* d_ws, size_t ws_size,
                              hipStream_t stream) {
    (void)in_sizes; (void)n_in; (void)d_ws; (void)ws_size; (void)out_size;
    const float* img  = (const float*)d_in[0];  // deshadow_img [2,3,384,384]
    const float* pp   = (const float*)d_in[1];  // pred_param   [2,6]
    const float* kern = (const float*)d_in[2];  // kernel       [2,486,384,384]
    float* out        = (float*)d_out;          // fuse_img ++ dark_img

    dim3 block(TX, TY, 1);
    dim3 grid(WW / TX, HH / TY, 2);
    hipLaunchKernelGGL(fuse_softmax_kernel, grid, block, 0, stream,
                       img, pp, kern, out);
}